// InteractionBlock_19172734009576
// MI455X (gfx1250) — compile-verified
//
#include <hip/hip_runtime.h>
#include <hip/hip_bf16.h>

#define N_NODES 10000
#define N_EDGES 100000
#define FCH 128
#define KCH 16
#define RCH 8
#define HCH 64
#define KF  (KCH * FCH)          // 2048
#define INV_AVG 0.1f             // 1 / AVG_NUM_NEIGHBORS

#define MT   5                   // m-tiles (16 rows each) per wave
#define MBLK (MT * 16)           // 80 rows per block; 10000 = 125 * 80
#define KC   32                  // K-chunk staged in LDS per double-buffer slot

// LDS slab geometry (floats), padded by the TDM for bank-conflict-free reads
#define BPITCH 136               // 128 + 8  (pad_interval=128DW, pad=8DW)
#define APITCH 36                // 32 + 4   (pad_interval=32DW,  pad=4DW)
#define BSLAB  (KC * BPITCH * 4)      // 17408 B
#define ASLAB  (MBLK * APITCH * 4)    // 11520 B
#define B0OFF  0
#define B1OFF  BSLAB
#define A0OFF  (2 * BSLAB)
#define A1OFF  (2 * BSLAB + ASLAB)
#define SMEMB  (2 * BSLAB + 2 * ASLAB)   // 57856 B dynamic LDS

// D# group1 word0 pad flags: pad_enable | pad_interval<<22 | pad_amount<<25
#define APAD ((1u << 20) | (4u << 22) | (3u << 25))   // every 32 DW, +4 DW
#define BPAD ((1u << 20) | (6u << 22) | (7u << 25))   // every 128 DW, +8 DW

typedef __attribute__((ext_vector_type(2))) float v2f;
typedef __attribute__((ext_vector_type(8))) float v8f;
typedef __attribute__((ext_vector_type(4))) unsigned int u32x4;
typedef __attribute__((ext_vector_type(8))) int i32x8;
typedef __attribute__((ext_vector_type(4))) int i32x4;

// ---------------------------------------------------------------------------
// Issue one 2D TDM tile load (global -> LDS).  D# layout per ISA §8.3/8.4.
// ---------------------------------------------------------------------------
__device__ __forceinline__ void tdm_load_2d(const void* gsrc, unsigned ldsOff,
                                            unsigned tensor_w, unsigned tensor_h,
                                            unsigned tile_w, unsigned tile_h,
                                            unsigned row_stride, unsigned padflags) {
    unsigned long long ga = (unsigned long long)gsrc;
    u32x4 g0;
    g0[0] = 1u;                                            // count=1, user D#
    g0[1] = ldsOff;                                        // lds_addr (bytes)
    g0[2] = (unsigned)(ga & 0xFFFFFFFFu);                  // global_addr[31:0]
    g0[3] = (unsigned)((ga >> 32) & 0x1FFFFFFu) | (2u << 30);  // [56:32] | type=2
    i32x8 g1;
    g1[0] = (int)((2u << 16) | padflags);                  // data_size=4B + pad cfg
    g1[1] = (int)((tensor_w & 0xFFFFu) << 16);             // tensor_dim0[15:0]
    g1[2] = (int)((tensor_w >> 16) | ((tensor_h & 0xFFFFu) << 16));
    g1[3] = (int)((tensor_h >> 16) | (tile_w << 16));      // tile_dim0
    g1[4] = (int)(tile_h & 0xFFFFu);                       // tile_dim1 (tile_dim2=0)
    g1[5] = (int)row_stride;                               // tensor_dim0_stride
    g1[6] = 0;
    g1[7] = 0;
    i32x4 z4 = {};
    i32x8 z8 = {};
    __builtin_amdgcn_tensor_load_to_lds(g0, g1, z4, z4, z8, 0);
}

// ---------------------------------------------------------------------------
// fp32 WMMA GEMM:  C[M x 128] = scale * (A[M x K] @ B[K x 128])
//  - block: 8 waves, MBLK(80) rows x 128 cols; wave w owns n-tile w, MT=5
//    accumulators along M (one B fragment feeds 5 WMMAs).
//  - Both A (80 x KC) and B (KC x 128) slabs staged in LDS by the TDM,
//    double-buffered: chunk c+1 DMA overlaps chunk c compute.
//  - TDM pad feature de-conflicts LDS banks (APITCH/BPITCH).
// Requires M % MBLK == 0 and K % KC == 0 (EXEC all-ones for WMMA).
// ---------------------------------------------------------------------------
__global__ void wmma_gemm_f32_tdm(const float* __restrict__ A,
                                  const float* __restrict__ B,
                                  float* __restrict__ C,
                                  int K, float scale) {
    extern __shared__ float smem[];

    const int lane = threadIdx.x & 31;
    const int wave = threadIdx.x >> 5;           // n-tile index 0..7
    const int mg   = blockIdx.x;                 // m-group (MBLK rows)
    const int l15  = lane & 15;
    const int nloc = wave * 16 + l15;            // column 0..127
    const int kp   = (lane >> 4) * 2;            // K sub-offset for half-wave

    const float* Ag = A + (size_t)mg * MBLK * K; // this block's A panel

    v8f acc[MT];
#pragma unroll
    for (int t = 0; t < MT; ++t) acc[t] = (v8f){};

    const int nchunks = K / KC;

    if (wave == 0) {                             // prologue: chunk 0 -> buf 0
        tdm_load_2d(B, B0OFF, FCH, KC, FCH, KC, FCH, BPAD);
        tdm_load_2d(Ag, A0OFF, (unsigned)K, MBLK, KC, MBLK, (unsigned)K, APAD);
    }

    for (int c = 0; c < nchunks; ++c) {
        const int cur = c & 1;
        if (wave == 0) {
            if (c + 1 < nchunks) {               // prefetch chunk c+1 -> buf cur^1
                const int nb = (c + 1) & 1;
                tdm_load_2d(B + (size_t)(c + 1) * KC * FCH,
                            nb ? B1OFF : B0OFF, FCH, KC, FCH, KC, FCH, BPAD);
                tdm_load_2d(Ag + (size_t)(c + 1) * KC,
                            nb ? A1OFF : A0OFF, (unsigned)K, MBLK, KC, MBLK,
                            (unsigned)K, APAD);
                __builtin_amdgcn_s_wait_tensorcnt(2);   // chunk c's pair done
            } else {
                __builtin_amdgcn_s_wait_tensorcnt(0);
            }
        }
        __syncthreads();                         // slab cur visible to all waves

        const float* Bs = smem + ((cur ? B1OFF : B0OFF) >> 2);
        const float* As = smem + ((cur ? A1OFF : A0OFF) >> 2);

#pragma unroll 2
        for (int kk = 0; kk < KC; kk += 4) {
            v2f b;
            b.x = Bs[(kk + kp)     * BPITCH + nloc];
            b.y = Bs[(kk + kp + 1) * BPITCH + nloc];
#pragma unroll
            for (int t = 0; t < MT; ++t) {
                const float* ap = As + (t * 16 + l15) * APITCH + kk + kp;
                v2f a;
                a.x = ap[0];
                a.y = ap[1];
                acc[t] = __builtin_amdgcn_wmma_f32_16x16x4_f32(
                    false, a, false, b, (short)0, acc[t], false, false);
            }
        }
        __syncthreads();                         // slab cur consumed (reused c+2)
    }

    // D layout: VGPR r -> row  mg*MBLK + t*16 + r + 8*(lane>>4),  col nloc
#pragma unroll
    for (int t = 0; t < MT; ++t) {
        const int mbase = mg * MBLK + t * 16 + 8 * (lane >> 4);
#pragma unroll
        for (int r = 0; r < 8; ++r)
            C[(size_t)(mbase + r) * FCH + nloc] = acc[t][r] * scale;
    }
}

// ---------------------------------------------------------------------------
// Radial MLP:  w = silu(re @ W_r1) @ W_r2 ;  ew[e,k] = mask*ef[e,k]*w[k]
// ---------------------------------------------------------------------------
__global__ void radial_edge_kernel(const float* __restrict__ re,
                                   const float* __restrict__ ef,
                                   const unsigned char* __restrict__ mask,
                                   const float* __restrict__ W1,
                                   const float* __restrict__ W2,
                                   float* __restrict__ ew) {
    const int e = blockIdx.x * blockDim.x + threadIdx.x;
    if (e >= N_EDGES) return;

    float r[RCH];
#pragma unroll
    for (int i = 0; i < RCH; ++i) r[i] = re[(size_t)e * RCH + i];

    float acc[KCH];
#pragma unroll
    for (int k = 0; k < KCH; ++k) acc[k] = 0.0f;

    for (int h = 0; h < HCH; ++h) {
        float hv = 0.0f;
#pragma unroll
        for (int i = 0; i < RCH; ++i) hv = fmaf(r[i], W1[i * HCH + h], hv);
        hv = hv / (1.0f + __expf(-hv));          // SiLU
#pragma unroll
        for (int k = 0; k < KCH; ++k) acc[k] = fmaf(hv, W2[h * KCH + k], acc[k]);
    }

    const float m = mask[e] ? 1.0f : 0.0f;
#pragma unroll
    for (int k = 0; k < KCH; ++k)
        ew[(size_t)e * KCH + k] = m * ef[(size_t)e * KCH + k] * acc[k];
}

// ---------------------------------------------------------------------------
// Rank-1 message scatter:  agg[recv, k*F + f] += ew[e,k] * x[send, f]
// 128 threads per edge (thread = f, coalesced); agg (82 MB) is L2-resident.
// ---------------------------------------------------------------------------
__global__ void scatter_msg_kernel(const float* __restrict__ ew,
                                   const float* __restrict__ X,
                                   const int* __restrict__ senders,
                                   const int* __restrict__ receivers,
                                   float* __restrict__ agg) {
    const int e = blockIdx.x * 2 + (threadIdx.x >> 7);   // 2 edges / block
    const int f = threadIdx.x & 127;

    const int s   = senders[e];
    const int rcv = receivers[e];
    const float xf = X[(size_t)s * FCH + f];

    const float* ewp  = ew + (size_t)e * KCH;
    float*       aggp = agg + (size_t)rcv * KF + f;

#pragma unroll
    for (int k = 0; k < KCH; ++k)
        atomicAdd(aggp + k * FCH, ewp[k] * xf);
}

// ---------------------------------------------------------------------------
extern "C" void kernel_launch(void* const* d_in, const int* in_sizes, int n_in,
                              void* d_out, int out_size, void* d_ws, size_t ws_size,
                              hipStream_t stream) {
    const float*         node_feats = (const float*)d_in[0];
    const float*         edge_feats = (const float*)d_in[1];
    const float*         radial     = (const float*)d_in[2];
    const int*           senders    = (const int*)d_in[3];
    const int*           receivers  = (const int*)d_in[4];
    const unsigned char* edge_mask  = (const unsigned char*)d_in[5];
    const float*         W_up       = (const float*)d_in[6];
    const float*         W_r1       = (const float*)d_in[7];
    const float*         W_r2       = (const float*)d_in[8];
    const float*         W_down     = (const float*)d_in[9];
    float*               out        = (float*)d_out;

    // Workspace: X | agg | ew  = 5.1 MB + 81.9 MB + 6.4 MB
    float* X   = (float*)d_ws;                       // [N_NODES, F]
    float* agg = X + (size_t)N_NODES * FCH;          // [N_NODES, K*F]
    float* ew  = agg + (size_t)N_NODES * KF;         // [N_EDGES, K]

    (void)hipMemsetAsync(agg, 0, (size_t)N_NODES * KF * sizeof(float), stream);

    // 1) X = node_feats @ W_up   (125 blocks x 80 rows, K=128)
    wmma_gemm_f32_tdm<<<N_NODES / MBLK, 256, SMEMB, stream>>>(
        node_feats, W_up, X, FCH, 1.0f);

    // 2) per-edge radial MLP + weighting + mask
    radial_edge_kernel<<<(N_EDGES + 255) / 256, 256, 0, stream>>>(
        radial, edge_feats, edge_mask, W_r1, W_r2, ew);

    // 3) rank-1 messages scattered into L2-resident agg
    scatter_msg_kernel<<<N_EDGES / 2, 256, 0, stream>>>(
        ew, X, senders, receivers, agg);

    // 4) out = (agg / avg_neighbors) @ W_down   (dominant 5.24 GFLOP GEMM)
    wmma_gemm_f32_tdm<<<N_NODES / MBLK, 256, SMEMB, stream>>>(
        agg, W_down, out, KF, INV_AVG);
}